// DNAS_DeformAxialDW_22840636080512
// MI455X (gfx1250) — compile-verified
//
#include <hip/hip_runtime.h>

// DNAS_DeformAxialDW on MI455X (gfx1250).
//
// out = x + conv_h(x) + conv_w(x): 7-tap depthwise, scalar fractional
// dilation r (bilinear between integer shifts, zero padding).
//
// Roofline: 51.4M px; HBM floor = 410 MB -> ~17.6us @ 23.3 TB/s. Essential
// math (28 folded FMA/px) is already ~HBM-time on the fp32 VALU, so the
// stencil is recast as two translation-invariant banded matmuls per plane:
//     OUT(16x16) = Sv(16xK) @ Xwin(Kx16) + Xwin(16xK) @ Sw(Kx16) + X
// (K = 16 + span, span ~ 6r+2), executed as chained V_WMMA_F32_16X16X4_F32.
// Per 256-px tile: ~14 WMMA + ~40 ds + small branchless VALU -> HBM-bound.
//
// Data movement: whole 224x224 fp32 plane staged into LDS (320KB/WGP) by the
// Tensor Data Mover in iterate mode (1 row/iteration, LDS pitch 225 floats;
// gcd(225,64)=1 => conflict-free column gathers). Band matrices Sv/Sw are
// built once per block in LDS while the TDM transfer is in flight.
//
// Guarded window loads are branchless: clamped address + cndmask-select,
// so EXEC stays all-ones (WMMA requirement) with no saveexec traffic, and
// a 2-stage software pipeline overlaps chunk q+1 loads with chunk q's WMMA.

typedef unsigned int v4u __attribute__((ext_vector_type(4)));
typedef int          v8i __attribute__((ext_vector_type(8)));
typedef int          v4i __attribute__((ext_vector_type(4)));
typedef float        v2f __attribute__((ext_vector_type(2)));
typedef float        v8f __attribute__((ext_vector_type(8)));

#define HW     224
#define PITCH  225                 // LDS row pitch (floats); odd => conflict-free
#define PLANE  (HW * HW)
#define KTAP   7
#define NTHR   1024
#define KCAP   64                  // max band K (exact for r <= ~7.8)
#define TILES  ((HW / 16) * (HW / 16))   // 196 full 16x16 tiles per plane

__global__ __launch_bounds__(NTHR) void deform_axial_dw_wmma_kernel(
    const float* __restrict__ x,
    const float* __restrict__ weight_h,   // (C,1,7,1) -> flat c*7+k
    const float* __restrict__ weight_w,   // (C,1,1,7) -> flat c*7+k
    const float* __restrict__ r_ptr,
    float* __restrict__ out,
    int C)
{
    extern __shared__ float smem[];
    float* s_x = smem;                       // PITCH*HW: plane, padded pitch
    float* s_S = smem + PITCH * HW;          // KCAP*16: vertical band, K-major
    float* s_W = s_S + KCAP * 16;            // KCAP*16: horizontal band, K-major

    const int tid   = threadIdx.x;
    const int plane = blockIdx.x;            // b*C + c
    const int c     = plane % C;
    const float* __restrict__ xp = x   + (size_t)plane * PLANE;
    float* __restrict__       op = out + (size_t)plane * PLANE;

    // ---------------- Stage plane into LDS (TDM, iterate mode) -------------
#if __has_builtin(__builtin_amdgcn_tensor_load_to_lds)
    if (tid < 32) {                          // wave-level DMA; one wave issues
        unsigned long long ga = (unsigned long long)(const void*)xp;
        v4u g0;                               // D# group 0 (ISA 8.3)
        g0[0] = 1u;                           // count=1 (valid, user mode)
        g0[1] = 0u;                           // lds_addr = 0 (s_x base)
        g0[2] = (unsigned)(ga & 0xFFFFFFFFu);
        g0[3] = (unsigned)((ga >> 32) & 0x01FFFFFFu) | (2u << 30);  // type=2
        v8i g1;                               // D# group 1 (ISA 8.4)
        g1[0] = (2 << 16) | (1 << 19);        // data_size=4B, iterate_enable
        g1[1] = (HW << 16);                   // tensor_dim0 @ bits[63:48]
        g1[2] = (HW << 16);                   // tensor_dim1 @ bits[95:80]
        g1[3] = (HW << 16);                   // tile_dim0 = 224
        g1[4] = 1;                            // tile_dim1 = 1 row / iteration
        g1[5] = HW;                           // tensor_dim0_stride
        g1[6] = 0;
        g1[7] = 0;
        v4i g2;                               // D# group 2 (iterate mode, 8.5)
        g2[0] = 1;                            // benign
        g2[1] = PITCH;                        // lds_addr_increment / iter
        g2[2] = HW;                           // global_addr_increment / iter
        g2[3] = ((HW - 1) << 16);             // iterate_count = 223 (224 rows)
        v4i g3; g3[0] = 0; g3[1] = 0; g3[2] = 0; g3[3] = 0;
#if __clang_major__ >= 23
        v8i gz = {0, 0, 0, 0, 0, 0, 0, 0};
        __builtin_amdgcn_tensor_load_to_lds(g0, g1, g2, g3, gz, 0);
#else
        __builtin_amdgcn_tensor_load_to_lds(g0, g1, g2, g3, 0);
#endif
    }
#else
    for (int i = tid; i < PLANE; i += NTHR) {
        int h = i / HW, w = i - h * HW;
        s_x[h * PITCH + w] = xp[i];
    }
#endif

    // ------------- Tap params + band matrices (overlaps TDM) ---------------
    float rv = r_ptr[0];
    rv = rv < 1.0f ? 1.0f : rv;               // jnp.clip(r, 1.0)
    int   d0[KTAP];
    float fr[KTAP];
#pragma unroll
    for (int k = 0; k < KTAP; ++k) {
        float delta = (float)(k - KTAP / 2) * rv;
        float df    = floorf(delta);
        d0[k] = (int)df;
        fr[k] = delta - df;
    }
    const int dmin = d0[0];                       // r>=1 => d0 increasing
    const int span = (d0[KTAP - 1] + 1) - dmin;
    int K = 16 + span;
    if (K > KCAP) K = KCAP;                       // exact for r <= ~7.8
    const int nk = (K + 3) >> 2;

    // S[q][i] = sum_k cA*[q==i+d0[k]-dmin] + cB*[q==i+d0[k]+1-dmin]
    if (tid < 16 * 4 * nk) {
        int q = tid >> 4, i = tid & 15;
        float sv = 0.0f, sw = 0.0f;
#pragma unroll
        for (int k = 0; k < KTAP; ++k) {
            float omf = 1.0f - fr[k];
            float whk = weight_h[c * KTAP + k];
            float wwk = weight_w[c * KTAP + k];
            int qa = i + d0[k] - dmin;
            if (q == qa)     { sv += whk * omf;   sw += wwk * omf;   }
            if (q == qa + 1) { sv += whk * fr[k]; sw += wwk * fr[k]; }
        }
        s_S[q * 16 + i] = sv;
        s_W[q * 16 + i] = sw;
    }

#if __has_builtin(__builtin_amdgcn_tensor_load_to_lds)
    if (tid < 32) __builtin_amdgcn_s_wait_tensorcnt(0);
#endif
    __syncthreads();

    // ---------------- WMMA tile loop ---------------------------------------
    const int lane = tid & 31;
    const int n    = lane & 15;      // tile col (B/C/D) or tile row m (A)
    const int half = lane >> 4;      // K-pair select in A/B layouts

    // Branchless guarded LDS read: clamp row/col, select 0 when OOB.
    auto ldRow = [&](int rr, int colByteBase) -> float {  // x[rr][col]
        int rc = rr < 0 ? 0 : (rr > HW - 1 ? HW - 1 : rr);
        float v = s_x[rc * PITCH + colByteBase];
        return ((unsigned)rr < (unsigned)HW) ? v : 0.0f;
    };
    auto ldCol = [&](int rowBase, int cc) -> float {      // x[row][cc]
        int ccl = cc < 0 ? 0 : (cc > HW - 1 ? HW - 1 : cc);
        float v = s_x[rowBase * PITCH + ccl];
        return ((unsigned)cc < (unsigned)HW) ? v : 0.0f;
    };

    for (int t = tid >> 5; t < TILES; t += NTHR / 32) {
        const int th = t / (HW / 16), tw = t - th * (HW / 16);
        const int h0 = th * 16, w0 = tw * 16;
        const int rbase = h0 + 8 * half;

        // C init = x tile (C/D layout: VGPR g -> row h0+g+8*half, col n)
        v8f acc;
#pragma unroll
        for (int g = 0; g < 8; ++g)
            acc[g] = s_x[(rbase + g) * PITCH + w0 + n];

#if __has_builtin(__builtin_amdgcn_wmma_f32_16x16x4_f32)
        // ---- Vertical: acc += Sv(16xK) @ Xwin(Kx16), 2-stage pipelined ----
        {
            v2f a, b;
            int kk = 2 * half;
            a[0] = s_S[(kk + 0) * 16 + n];
            a[1] = s_S[(kk + 1) * 16 + n];
            b[0] = ldRow(h0 + dmin + kk,     w0 + n);
            b[1] = ldRow(h0 + dmin + kk + 1, w0 + n);
            for (int q = 0; q < nk; ++q) {
                const int qn = (q + 1 < nk) ? q + 1 : q;   // preload next chunk
                const int kn = qn * 4 + 2 * half;
                v2f an, bn;
                an[0] = s_S[(kn + 0) * 16 + n];
                an[1] = s_S[(kn + 1) * 16 + n];
                bn[0] = ldRow(h0 + dmin + kn,     w0 + n);
                bn[1] = ldRow(h0 + dmin + kn + 1, w0 + n);
                acc = __builtin_amdgcn_wmma_f32_16x16x4_f32(
                    false, a, false, b, (short)0, acc, false, false);
                a = an; b = bn;
            }
        }
        // ---- Horizontal: acc += Xwin(16xK) @ Sw(Kx16), 2-stage pipelined --
        {
            v2f a, b;
            int kk = 2 * half;
            a[0] = ldCol(h0 + n, w0 + dmin + kk);
            a[1] = ldCol(h0 + n, w0 + dmin + kk + 1);
            b[0] = s_W[(kk + 0) * 16 + n];
            b[1] = s_W[(kk + 1) * 16 + n];
            for (int q = 0; q < nk; ++q) {
                const int qn = (q + 1 < nk) ? q + 1 : q;
                const int kn = qn * 4 + 2 * half;
                v2f an, bn;
                an[0] = ldCol(h0 + n, w0 + dmin + kn);
                an[1] = ldCol(h0 + n, w0 + dmin + kn + 1);
                bn[0] = s_W[(kn + 0) * 16 + n];
                bn[1] = s_W[(kn + 1) * 16 + n];
                acc = __builtin_amdgcn_wmma_f32_16x16x4_f32(
                    false, a, false, b, (short)0, acc, false, false);
                a = an; b = bn;
            }
        }
#else
        // Fallback: folded-coefficient scalar path (2 FMA per tap per axis)
        for (int g = 0; g < 8; ++g) {
            const int hh = rbase + g, wc = w0 + n;
            float v = acc[g];
            for (int k = 0; k < KTAP; ++k) {
                float omf = 1.0f - fr[k];
                float whk = weight_h[c * KTAP + k], wwk = weight_w[c * KTAP + k];
                int r0 = hh + d0[k], c0 = wc + d0[k];
                v += whk * (omf * ldRow(r0, wc) + fr[k] * ldRow(r0 + 1, wc))
                   + wwk * (omf * ldCol(hh, c0) + fr[k] * ldCol(hh, c0 + 1));
            }
            acc[g] = v;
        }
#endif
        // Store D (coalesced per 16-lane half)
#pragma unroll
        for (int g = 0; g < 8; ++g)
            op[(rbase + g) * HW + w0 + n] = acc[g];
    }
}

extern "C" void kernel_launch(void* const* d_in, const int* in_sizes, int n_in,
                              void* d_out, int out_size, void* d_ws, size_t ws_size,
                              hipStream_t stream) {
    (void)n_in; (void)d_ws; (void)ws_size;
    const float* x  = (const float*)d_in[0];
    const float* wh = (const float*)d_in[1];
    const float* ww = (const float*)d_in[2];
    const float* r  = (const float*)d_in[3];
    float* out      = (float*)d_out;

    const int C      = in_sizes[1] / KTAP;
    const int planes = out_size / PLANE;

    const size_t shmem = (size_t)(PITCH * HW + 2 * KCAP * 16) * sizeof(float);
    deform_axial_dw_wmma_kernel<<<dim3(planes), dim3(NTHR), shmem, stream>>>(
        x, wh, ww, r, out, C);
}